// PartialMatchingLoss_64991445123087
// MI455X (gfx1250) — compile-verified
//
#include <hip/hip_runtime.h>

typedef __attribute__((ext_vector_type(2))) float v2f;
typedef __attribute__((ext_vector_type(8))) float v8f;

#define NBATCH 8
#define NC 8192
#define NP 2048
#define THRESH 0.05f

__global__ void pml_init(float* acc) {
    acc[0] = 0.0f;
    acc[1] = 0.0f;
}

__global__ __launch_bounds__(256) void pml_chamfer(const float* __restrict__ comp,
                                                   const float* __restrict__ part,
                                                   float* __restrict__ acc) {
    __shared__ float lds_pts[NP * 4];   // (x,y,z,0) per partial point
    __shared__ float lds_b2[NP];        // |b|^2 per partial point

    const int tid   = threadIdx.x;
    const int batch = blockIdx.y;

    // Stage this batch's partial cloud into LDS, padded to 4 floats.
    const float* pb = part + (size_t)batch * NP * 3;
    for (int i = tid; i < NP; i += 256) {
        float x = pb[i * 3 + 0];
        float y = pb[i * 3 + 1];
        float z = pb[i * 3 + 2];
        lds_pts[i * 4 + 0] = x;
        lds_pts[i * 4 + 1] = y;
        lds_pts[i * 4 + 2] = z;
        lds_pts[i * 4 + 3] = 0.0f;
        lds_b2[i] = x * x + y * y + z * z;
    }
    __syncthreads();

    const int wave = tid >> 5;
    const int lane = tid & 31;
    const int half = lane >> 4;     // 0: K={0,1}, 1: K={2,3}
    const int lm   = lane & 15;

    // A-matrix tile: 16 completed points (rows), K=4 padded coords.
    const int m = (blockIdx.x * 8 + wave) * 16 + lm;
    const float* cp = comp + ((size_t)batch * NC + m) * 3;
    const float x = cp[0], y = cp[1], z = cp[2];
    const float a2 = x * x + y * y + z * z;

    v2f a;
    a.x = half ? z : x;
    a.y = half ? 0.0f : y;

    // Seed the WMMA accumulator with -0.5*a2[M] so that
    //   D = A.B + C = dot - 0.5*a2[M]   and   d2 = b2 - 2*D.
    // C layout: lane l, slot v holds (M = v + 8*half, N = l%16); a2 of point p lives in lane p.
    v8f a2c;
#pragma unroll
    for (int v = 0; v < 8; ++v)
        a2c[v] = -0.5f * __shfl(a2, v + 8 * half, 32);

    float best[8];
#pragma unroll
    for (int v = 0; v < 8; ++v)
        best[v] = 3.4e38f;

    const float2* lp = (const float2*)lds_pts;

#pragma unroll 4
    for (int nb = 0; nb < NP; nb += 16) {
        const int n = nb + lm;
        // B-matrix tile (4x16): same per-lane layout as A, mirrored to columns.
        float2 bv = lp[n * 2 + half];
        v2f bb;
        bb.x = bv.x;
        bb.y = bv.y;
        const float b2 = lds_b2[n];

        // D[m][n] = dot(a_m, b_n) - 0.5*a2[m]   (16x16 tile, K=4)
        v8f c = __builtin_amdgcn_wmma_f32_16x16x4_f32(false, a, false, bb,
                                                      (short)0, a2c, false, false);

        // d2 = a2 + b2 - 2*dot; clamp-to-0 is monotone, hoisted out of the loop.
#pragma unroll
        for (int v = 0; v < 8; ++v)
            best[v] = fminf(best[v], fmaf(-2.0f, c[v], b2));
    }

    // Min across the 16 lanes of each half: completes the reduction over all N.
#pragma unroll
    for (int mask = 1; mask < 16; mask <<= 1) {
#pragma unroll
        for (int v = 0; v < 8; ++v)
            best[v] = fminf(best[v], __shfl_xor(best[v], mask, 32));
    }

    // Lane lm==0 of each half holds mins for 8 distinct completed points.
    if (lm == 0) {
        float s = 0.0f, cnt = 0.0f;
#pragma unroll
        for (int v = 0; v < 8; ++v) {
            float d2 = fmaxf(best[v], 0.0f);    // deferred clamp
            if (d2 < THRESH) {
                s += d2;
                cnt += 1.0f;
            }
        }
        atomicAdd(&acc[0], s);
        atomicAdd(&acc[1], cnt);
    }
}

__global__ void pml_finalize(const float* __restrict__ acc, float* __restrict__ out) {
    const float s = acc[0];
    const float c = acc[1];
    out[0] = (c > 0.0f) ? (s / (c + 1e-6f)) : 0.0f;
}

extern "C" void kernel_launch(void* const* d_in, const int* in_sizes, int n_in,
                              void* d_out, int out_size, void* d_ws, size_t ws_size,
                              hipStream_t stream) {
    const float* comp = (const float*)d_in[0];  // (8, 8192, 3) f32
    const float* part = (const float*)d_in[1];  // (8, 2048, 3) f32
    float* out = (float*)d_out;                 // scalar loss
    float* acc = (float*)d_ws;                  // [sum, count]

    pml_init<<<1, 1, 0, stream>>>(acc);
    dim3 grid(NC / (16 * 8), NBATCH);           // 64 x 8 blocks, 8 waves each
    pml_chamfer<<<grid, 256, 0, stream>>>(comp, part, acc);
    pml_finalize<<<1, 1, 0, stream>>>(acc, out);
}